// lct_fk_26525718020682
// MI455X (gfx1250) — compile-verified
//
#include <hip/hip_runtime.h>

// ---------------------------------------------------------------------------
// f-k migration (LCT) pipeline for MI455X / gfx1250 (wave32, WMMA).
//   prep  -> pad+scale+sqrt into complex 256^3
//   3x fft256_pass (forward, one per dim, in-place, WMMA-based radix-16^2)
//   stolt -> trilinear Stolt resample (both fftshifts folded into indices)
//   3x fft256_pass (inverse)
//   crop  -> real part of [:128,:128,:128], * 1/256^3
// Workspace: two complex volumes of 2 x 256^3 float2  (2 x 268 MB).
// ---------------------------------------------------------------------------

typedef float v2f __attribute__((ext_vector_type(2)));
typedef float v8f __attribute__((ext_vector_type(8)));

#define VOLN   (1 << 24)          /* 256^3 */
#define PI_F   3.14159265358979323846f

__device__ __forceinline__ v8f wmma4(v2f a, v2f b, v8f c) {
    // D = A(16x4,f32) * B(4x16,f32) + C(16x16,f32)   -- v_wmma_f32_16x16x4_f32
    return __builtin_amdgcn_wmma_f32_16x16x4_f32(false, a, false, b,
                                                 (short)0, c, false, false);
}

// ---------------------------------------------------------------------------
// Prep: data = sqrt(relu(f * (z/(M-1))^2)), zero-padded into 256^3 complex.
// ---------------------------------------------------------------------------
__global__ __launch_bounds__(256)
void prep_kernel(const float* __restrict__ in, float2* __restrict__ A)
{
    size_t gid = (size_t)blockIdx.x * 256 + threadIdx.x;   // < 2*VOLN
    int volId = (int)(gid >> 24);
    int r = (int)(gid & (VOLN - 1));
    int z = r >> 16, y = (r >> 8) & 255, x = r & 255;
    float val = 0.0f;
    if (z < 128 && y < 128 && x < 128) {
        float v = in[(((size_t)volId * 128 + z) * 128 + y) * 128 + x];
        float g = (float)z * (1.0f / 127.0f);
        float s = v * g * g;
        val = (s > 0.0f) ? sqrtf(s) : 0.0f;
    }
    A[gid] = make_float2(val, 0.0f);
}

// ---------------------------------------------------------------------------
// In-place 256-point line FFT along dimension DIM (0=x,1=y,2=z).
// SGN = -1 forward, +1 inverse (normalization folded into crop).
// 256 = 16*16 four-step:  T = F16 * Xm ; T' = T .* W256 ; U = T' * F16
// with X[k1 + 16*k2] = U[k1][k2].  Each 16x16x16 complex matmul is
// 4 K-chunks x 4 real v_wmma_f32_16x16x4_f32.
// One wave per line; block of 8 waves handles an 8-line coalesced tile.
// ---------------------------------------------------------------------------
template<int DIM, int SGN>
__global__ __launch_bounds__(256)
void fft256_pass(float2* __restrict__ buf)
{
    __shared__ float2 sF[256];        // F16[m][k]   = exp(i*sgn*2pi*m*k/16)
    __shared__ float2 sTw[256];       // W256[m][n]  = exp(i*sgn*2pi*m*n/256)
    __shared__ float2 sLine[8][260];  // 8 lines of 256 (padded)
    __shared__ float2 sScr[8][256];   // per-wave transpose scratch

    const int tid = threadIdx.x;
    const float sgn = (float)SGN;

    {   // build DFT + twiddle tables (one element per thread)
        int m = tid >> 4, k = tid & 15;
        float s, c;
        float a16 = sgn * (2.0f * PI_F / 16.0f) * (float)(m * k);
        __sincosf(a16, &s, &c);   sF[tid]  = make_float2(c, s);
        float a256 = sgn * (2.0f * PI_F / 256.0f) * (float)(m * k);
        __sincosf(a256, &s, &c);  sTw[tid] = make_float2(c, s);
    }

    // ---- tile addressing -------------------------------------------------
    const int tileId = blockIdx.x;          // 2 vols * 8192 tiles
    const int volId  = tileId >> 13;
    const int lam0   = (tileId & 8191) << 3;  // first of 8 lines
    const size_t volBase = (size_t)volId * (size_t)VOLN;
    size_t base, sL, sI;
    if (DIM == 0)      { base = (size_t)lam0 * 256;                          sL = 256; sI = 1;     }
    else if (DIM == 1) { base = ((size_t)(lam0 >> 8)) * 65536 + (lam0 & 255); sL = 1;  sI = 256;   }
    else               { base = (size_t)lam0;                                 sL = 1;  sI = 65536; }

    __syncthreads();

    // ---- coalesced tile load --------------------------------------------
    #pragma unroll
    for (int c = 0; c < 8; ++c) {
        int e = c * 256 + tid;
        int l, i;
        if (DIM == 0) { l = e >> 8; i = e & 255; }   // lines contiguous
        else          { l = e & 7;  i = e >> 3;  }   // x-adjacent lines
        sLine[l][i] = buf[volBase + base + (size_t)l * sL + (size_t)i * sI];
    }
    __syncthreads();

    // ---- per-wave 256-pt FFT --------------------------------------------
    const int w  = tid >> 5;          // wave id == line slot
    const int ln = tid & 15;          // N / M index within fragment
    const int lh = (tid >> 4) & 1;    // half-wave

    v8f zero = {0,0,0,0,0,0,0,0};
    v8f cr = zero, ci = zero;

    // Step A:  T = F16 * Xm,  Xm[n1][n2] = x[16*n1 + n2]
    #pragma unroll
    for (int kc = 0; kc < 4; ++kc) {
        int K0 = kc * 4 + 2 * lh;
        int K1 = K0 + 1;
        float2 f0 = sF[ln * 16 + K0];
        float2 f1 = sF[ln * 16 + K1];
        v2f are  = {  f0.x,  f1.x };
        v2f aim  = {  f0.y,  f1.y };
        v2f aimn = { -f0.y, -f1.y };
        float2 x0 = sLine[w][K0 * 16 + ln];
        float2 x1 = sLine[w][K1 * 16 + ln];
        v2f bre = { x0.x, x1.x };
        v2f bim = { x0.y, x1.y };
        cr = wmma4(are,  bre, cr);
        cr = wmma4(aimn, bim, cr);   // re -= Fim*Xim
        ci = wmma4(are,  bim, ci);
        ci = wmma4(aim,  bre, ci);
    }

    // Twiddle T' = T .* W256^{k1*n2}, restage to scratch for transpose
    #pragma unroll
    for (int v = 0; v < 8; ++v) {
        int m = v + 8 * lh;                 // k1 row
        float2 tw = sTw[m * 16 + ln];
        float tr = cr[v] * tw.x - ci[v] * tw.y;
        float ti = cr[v] * tw.y + ci[v] * tw.x;
        sScr[w][m * 16 + ln] = make_float2(tr, ti);
    }
    __syncthreads();

    // Step C:  U = T' * F16
    cr = zero; ci = zero;
    #pragma unroll
    for (int kc = 0; kc < 4; ++kc) {
        int K0 = kc * 4 + 2 * lh;
        int K1 = K0 + 1;
        float2 t0 = sScr[w][ln * 16 + K0];
        float2 t1 = sScr[w][ln * 16 + K1];
        v2f are  = {  t0.x,  t1.x };
        v2f aim  = {  t0.y,  t1.y };
        v2f aimn = { -t0.y, -t1.y };
        float2 g0 = sF[K0 * 16 + ln];
        float2 g1 = sF[K1 * 16 + ln];
        v2f bre = { g0.x, g1.x };
        v2f bim = { g0.y, g1.y };
        cr = wmma4(are,  bre, cr);
        cr = wmma4(aimn, bim, cr);
        ci = wmma4(are,  bim, ci);
        ci = wmma4(aim,  bre, ci);
    }

    // Digit-reversed writeback: X[k1 + 16*k2] = U[k1][k2]
    #pragma unroll
    for (int v = 0; v < 8; ++v) {
        int m = v + 8 * lh;                 // k1
        sLine[w][m + 16 * ln] = make_float2(cr[v], ci[v]);
    }
    __syncthreads();

    // ---- coalesced tile store -------------------------------------------
    #pragma unroll
    for (int c = 0; c < 8; ++c) {
        int e = c * 256 + tid;
        int l, i;
        if (DIM == 0) { l = e >> 8; i = e & 255; }
        else          { l = e & 7;  i = e >> 3;  }
        buf[volBase + base + (size_t)l * sL + (size_t)i * sI] = sLine[l][i];
    }
}

// ---------------------------------------------------------------------------
// Stolt resample. Both fftshifts folded in:
//   output index j (IFFT input layout)  ->  samp index  = (j+128)&255
//   fre index i (shifted spectrum)      ->  FFT buffer  = (i+128)&255
// py = y-0.5, px = x-0.5 (exact half-weights); pz from dispersion relation.
// ---------------------------------------------------------------------------
__global__ __launch_bounds__(256)
void stolt_kernel(const float2* __restrict__ F, float2* __restrict__ B)
{
    size_t gid = (size_t)blockIdx.x * 256 + threadIdx.x;   // < 2*VOLN
    int volId = (int)(gid >> 24);
    int r = (int)(gid & (VOLN - 1));
    int jz = r >> 16, jy = (r >> 8) & 255, jx = r & 255;
    int z = (jz + 128) & 255;      // pre-2nd-shift samp coordinates
    int y = (jy + 128) & 255;
    int x = (jx + 128) & 255;

    float2 acc = make_float2(0.0f, 0.0f);
    if (z >= 129) {                               // z-mask (z0pos = M+1)
        float gz = (float)(z - 128) * (1.0f / 128.0f);
        float gy = (float)(y - 128) * (1.0f / 128.0f);
        float gx = (float)(x - 128) * (1.0f / 128.0f);
        // fk = (N*trange/(M*width*4))^2 = 0.32^2
        float gzn = sqrtf(0.1024f * (gx * gx + gy * gy) + gz * gz);
        float pz = (gzn + 1.0f) * 128.0f - 0.5f;
        int   z0 = (int)floorf(pz);
        float dz = pz - floorf(pz);
        int   y0 = y - 1, x0 = x - 1;             // dy = dx = 0.5 exactly
        const float2* Fv = F + (size_t)volId * VOLN;

        float wz[2] = { 1.0f - dz, dz };
        #pragma unroll
        for (int a = 0; a < 2; ++a) {
            int iz = z0 + a;
            if (iz < 0 || iz > 255) continue;
            size_t zoff = (size_t)((iz + 128) & 255) << 16;
            #pragma unroll
            for (int bb = 0; bb < 2; ++bb) {
                int iy = y0 + bb;
                if (iy < 0 || iy > 255) continue;
                size_t yoff = (size_t)((iy + 128) & 255) << 8;
                #pragma unroll
                for (int cc = 0; cc < 2; ++cc) {
                    int ix = x0 + cc;
                    if (ix < 0 || ix > 255) continue;
                    float2 v = Fv[zoff + yoff + (size_t)((ix + 128) & 255)];
                    float wt = wz[a] * 0.25f;
                    acc.x += wt * v.x;
                    acc.y += wt * v.y;
                }
            }
        }
        float scale = fabsf(gz) / (gzn + 1e-8f);
        acc.x *= scale;
        acc.y *= scale;
    }
    B[gid] = acc;
}

// ---------------------------------------------------------------------------
// Crop real part of [:128,:128,:128]; fold the 1/256^3 IFFT normalization.
// ---------------------------------------------------------------------------
__global__ __launch_bounds__(256)
void crop_kernel(const float2* __restrict__ B, float* __restrict__ out)
{
    size_t gid = (size_t)blockIdx.x * 256 + threadIdx.x;   // < 2*128^3
    int volId = (int)(gid >> 21);
    int r = (int)(gid & ((1 << 21) - 1));
    int z = r >> 14, y = (r >> 7) & 127, x = r & 127;
    float2 v = B[(size_t)volId * VOLN + ((size_t)z << 16) + (y << 8) + x];
    out[gid] = v.x * (1.0f / 16777216.0f);
}

// ---------------------------------------------------------------------------
extern "C" void kernel_launch(void* const* d_in, const int* in_sizes, int n_in,
                              void* d_out, int out_size, void* d_ws, size_t ws_size,
                              hipStream_t stream)
{
    const float* in = (const float*)d_in[0];
    float*      out = (float*)d_out;
    float2* A = (float2*)d_ws;                  // 2 x 256^3 complex (268 MB)
    float2* B = A + (size_t)2 * VOLN;           // 2 x 256^3 complex (268 MB)

    prep_kernel<<<131072, 256, 0, stream>>>(in, A);

    fft256_pass<0, -1><<<16384, 256, 0, stream>>>(A);
    fft256_pass<1, -1><<<16384, 256, 0, stream>>>(A);
    fft256_pass<2, -1><<<16384, 256, 0, stream>>>(A);

    stolt_kernel<<<131072, 256, 0, stream>>>(A, B);

    fft256_pass<0, 1><<<16384, 256, 0, stream>>>(B);
    fft256_pass<1, 1><<<16384, 256, 0, stream>>>(B);
    fft256_pass<2, 1><<<16384, 256, 0, stream>>>(B);

    crop_kernel<<<16384, 256, 0, stream>>>(B, out);
}